// SimplifiedBioelectricCore_22454089023958
// MI455X (gfx1250) — compile-verified
//
#include <hip/hip_runtime.h>
#include <hip/hip_bf16.h>

// Bandwidth-bound 3x3 stencil + elementwise pipeline for MI455X (gfx1250).
// Memory floor: ~512MB / 23.3TB/s ~= 22us. Strategy: one fused pass,
// async global->LDS staging of the voltage tile (CDNA5 ASYNCcnt path),
// v_tanh_f32 transcendentals, non-temporal streaming for the 5 outputs.

#define FD 4096
constexpr int TILE     = 64;                  // 64x64 outputs per block
constexpr int HALO_X   = 4;                   // 16B-aligned side halo (floats)
constexpr int LDSW     = TILE + 2 * HALO_X;   // 72 floats per LDS row
constexpr int LDSH     = TILE + 2;            // 66 LDS rows (1-row halo)
constexpr int CHUNKS_X = LDSW / 4;            // 18 float4 chunks per row
constexpr int NCHUNK   = LDSH * CHUNKS_X;     // 1188 async B128 transfers
constexpr int NTHREADS = 256;                 // 8 waves (wave32)

// Native clang vector type: required by __builtin_nontemporal_load/store
// (HIP's float4 is a struct and is rejected).
typedef __attribute__((ext_vector_type(4))) float f32x4;

__device__ __forceinline__ float fast_tanh(float x) {
#if __has_builtin(__builtin_amdgcn_tanhf)
  return __builtin_amdgcn_tanhf(x);
#elif __has_builtin(__builtin_amdgcn_tanh_f32)
  return __builtin_amdgcn_tanh_f32(x);
#else
  return tanhf(x);
#endif
}

// Byte offset of a __shared__ object inside LDS (addrspace(3) ptrtoint).
__device__ __forceinline__ unsigned lds_off(void* p) {
  return (unsigned)(unsigned long long)(__attribute__((address_space(3))) void*)p;
}

__device__ __forceinline__ void wait_asynccnt0() {
#if __has_builtin(__builtin_amdgcn_s_wait_asynccnt)
  __builtin_amdgcn_s_wait_asynccnt(0);
#else
  asm volatile("s_wait_asynccnt 0" ::: "memory");
#endif
}

__global__ __launch_bounds__(NTHREADS)
void bioelectric_fused_kernel(const float* __restrict__ voltage,
                              const float* __restrict__ calcium_gradient,
                              const float* __restrict__ morph,
                              float* __restrict__ out) {
  __shared__ float tile[LDSH * LDSW];  // 19,008 B of 320KB/WGP

  const int t   = threadIdx.x;
  const int gx0 = blockIdx.x * TILE;
  const int gy0 = blockIdx.y * TILE;

  // ---- Phase 1: async-stage voltage tile (+halo) into LDS ----------------
  // Each lane issues GLOBAL_LOAD_ASYNC_TO_LDS_B128: 16B memory -> 16B LDS,
  // tracked by ASYNCcnt (no VGPR data, no LOADcnt pressure).
  const unsigned lbase = lds_off((void*)tile);
  for (int i = t; i < NCHUNK; i += NTHREADS) {
    const int r = i / CHUNKS_X;
    const int c = i - r * CHUNKS_X;
    int gy = gy0 - 1 + r;
    gy = gy < 0 ? 0 : (gy > FD - 1 ? FD - 1 : gy);          // clamp (safe: see below)
    int gxb = gx0 - HALO_X + c * 4;
    gxb = gxb < 0 ? 0 : (gxb > FD - 4 ? FD - 4 : gxb);       // keep 16B chunk in-bounds
    const unsigned goff = (unsigned)((gy * FD + gxb) * 4);   // byte offset, 16B aligned
    const unsigned loff = lbase + (unsigned)((r * LDSW + c * 4) * 4);
    asm volatile("global_load_async_to_lds_b128 %0, %1, %2"
                 :: "v"(loff), "v"(goff), "s"(voltage)
                 : "memory");
  }
  wait_asynccnt0();      // own wave's transfers landed in LDS
  __syncthreads();       // all 8 waves' transfers visible

  // Clamped halo cells hold duplicated edge data; they are only ever read as
  // neighbors of *boundary* outputs, which select the center value (no conv),
  // so the clamp never leaks into a result.

  // ---- Phase 2: stencil + activations, streamed outputs ------------------
  const int tx = t & 15;           // 16 threads across x, 4 outputs each
  const int ty = t >> 4;           // 16 threads across y, 4 rows each (stride 16)
  const int lx = HALO_X + tx * 4;  // LDS col of first output
  const int gx = gx0 + tx * 4;
  const unsigned long long N = (unsigned long long)FD * FD;

#pragma unroll
  for (int rr = 0; rr < 4; ++rr) {
    const int ry = ty + rr * 16;
    const int gy = gy0 + ry;
    const int lr = 1 + ry;
    const float* row0 = &tile[(lr - 1) * LDSW];
    const float* row1 = &tile[lr * LDSW];
    const float* row2 = &tile[(lr + 1) * LDSW];

    float r0[6], r1[6], r2[6];
#pragma unroll
    for (int k = 0; k < 6; ++k) {
      r0[k] = row0[lx - 1 + k];
      r1[k] = row1[lx - 1 + k];
      r2[k] = row2[lx - 1 + k];
    }

    const bool yedge = (gy == 0) | (gy == FD - 1);
    float nv[4], sg[4];
#pragma unroll
    for (int j = 0; j < 4; ++j) {
      const int x = gx + j;
      const float conv =
          0.05f * ((r0[j] + r0[j + 2]) + (r2[j] + r2[j + 2])) +
          0.10f * ((r0[j + 1] + r2[j + 1]) + (r1[j] + r1[j + 2])) +
          0.40f * r1[j + 1];
      const bool edge = yedge | (x == 0) | (x == FD - 1);
      const float v = edge ? r1[j + 1] : conv;           // boundary keeps raw voltage
      const float z = fast_tanh(__builtin_fmaf(0.9f, v, -0.02f));
      nv[j] = z;
      sg[j] = __builtin_fmaf(0.5f, fast_tanh(0.5f * z), 0.5f);  // exact sigmoid
    }

    const unsigned long long idx = (unsigned long long)gy * FD + gx;
    const f32x4 cgv = __builtin_nontemporal_load((const f32x4*)(calcium_gradient + idx));
    const f32x4 mo  = __builtin_nontemporal_load((const f32x4*)(morph + idx));

    f32x4 o0 = {nv[0], nv[1], nv[2], nv[3]};
    f32x4 o1 = {sg[0], sg[1], sg[2], sg[3]};
    f32x4 o2 = {1.0f - sg[0], 1.0f - sg[1], 1.0f - sg[2], 1.0f - sg[3]};
    f32x4 o3 = {__builtin_fmaf(0.9f, cgv.x, 0.1f * sg[0]),
                __builtin_fmaf(0.9f, cgv.y, 0.1f * sg[1]),
                __builtin_fmaf(0.9f, cgv.z, 0.1f * sg[2]),
                __builtin_fmaf(0.9f, cgv.w, 0.1f * sg[3])};
    f32x4 o4 = {__builtin_fmaf(0.95f, mo.x, 0.05f * nv[0]),
                __builtin_fmaf(0.95f, mo.y, 0.05f * nv[1]),
                __builtin_fmaf(0.95f, mo.z, 0.05f * nv[2]),
                __builtin_fmaf(0.95f, mo.w, 0.05f * nv[3])};

    __builtin_nontemporal_store(o0, (f32x4*)(out + 0 * N + idx));  // new_voltage
    __builtin_nontemporal_store(o1, (f32x4*)(out + 1 * N + idx));  // sodium
    __builtin_nontemporal_store(o2, (f32x4*)(out + 2 * N + idx));  // potassium
    __builtin_nontemporal_store(o3, (f32x4*)(out + 3 * N + idx));  // calcium
    __builtin_nontemporal_store(o4, (f32x4*)(out + 4 * N + idx));  // new_morph
  }
}

extern "C" void kernel_launch(void* const* d_in, const int* in_sizes, int n_in,
                              void* d_out, int out_size, void* d_ws, size_t ws_size,
                              hipStream_t stream) {
  const float* voltage = (const float*)d_in[0];
  const float* calcium = (const float*)d_in[1];
  const float* morph   = (const float*)d_in[2];
  float* out = (float*)d_out;

  dim3 grid(FD / TILE, FD / TILE);   // 64 x 64 blocks
  dim3 block(NTHREADS);
  bioelectric_fused_kernel<<<grid, block, 0, stream>>>(voltage, calcium, morph, out);
}